// Embedder_23957327577748
// MI455X (gfx1250) — compile-verified
//
#include <hip/hip_runtime.h>

// ---------------- problem constants ----------------
#define RB     128            // heatmap resolution
#define BBASE  64             // batch of base heatmaps
#define KCLIP  100            // boxes per base map
#define NB     (BBASE*KCLIP)  // 6400 samples
#define NCH    3              // conv channels
#define PO     29             // pooled spatial
#define QO     27             // conv2 output spatial
#define FDIM   2187           // 3*27*27
#define KP     2192           // FDIM padded to /16 (zero filled)
#define NOUT   512
#define EPSBN  1e-5f
#define NEGINF (-3.402823466e38f)

typedef __attribute__((ext_vector_type(2))) float v2f;
typedef __attribute__((ext_vector_type(8))) float v8f;

// ---------------- workspace layout (floats) ----------------
#define O_STATS1 0
#define O_STATS2 8
#define O_BEFF   16
#define O_POOLED 1024
#define O_ACT    (O_POOLED + (size_t)NB*NCH*PO*PO)
#define O_WEFF   (O_ACT + (size_t)NB*KP)

// ---------------- kernel 0: zero the stat accumulators ----------------
__global__ void k_init(float* ws) {
  if (threadIdx.x < 16) ws[threadIdx.x] = 0.0f;
}

// ---------------- kernel 1: BN1 batch statistics ----------------
__global__ __launch_bounds__(256)
void k_stats1(const float* __restrict__ hm, const float* __restrict__ dets,
              const float* __restrict__ w1, const float* __restrict__ b1,
              float* __restrict__ stats) {
  const int n = blockIdx.x, t = threadIdx.x;
  const float* H = hm + (size_t)(n / KCLIP) * (RB * RB);
  const float d0 = dets[n*6+0], d1 = dets[n*6+1], d2 = dets[n*6+2], d3 = dets[n*6+3];
  const int left = (int)d0, up = (int)d1, right = (int)(d0 + d2), low = (int)(d1 + d3);
  float W[3][9];
  #pragma unroll
  for (int oc = 0; oc < 3; ++oc)
    #pragma unroll
    for (int k = 0; k < 9; ++k) W[oc][k] = w1[oc*9+k];
  const float cb0 = b1[0], cb1 = b1[1], cb2 = b1[2];
  float s0=0,s1=0,s2=0, q0=0,q1=0,q2=0;
  for (int i = 0; i < (RB*RB)/256; ++i) {
    const int p = i*256 + t, r = p >> 7, c = p & 127;
    float a0 = cb0, a1 = cb1, a2 = cb2;
    #pragma unroll
    for (int dr = 0; dr < 3; ++dr) {
      const int rr = r + dr - 1;
      const bool rok = (rr >= 0) && (rr < RB) && (rr >= up) && (rr < low);
      #pragma unroll
      for (int dc = 0; dc < 3; ++dc) {
        const int cc = c + dc - 1;
        float v = 0.0f;
        if (rok && cc >= 0 && cc < RB && cc >= left && cc < right) v = H[rr*RB+cc];
        a0 = fmaf(W[0][dr*3+dc], v, a0);
        a1 = fmaf(W[1][dr*3+dc], v, a1);
        a2 = fmaf(W[2][dr*3+dc], v, a2);
      }
    }
    a0 = fmaxf(a0, 0.f); a1 = fmaxf(a1, 0.f); a2 = fmaxf(a2, 0.f);
    s0 += a0; q0 += a0*a0; s1 += a1; q1 += a1*a1; s2 += a2; q2 += a2*a2;
  }
  __shared__ float rs[256];
  float vals[6] = {s0, s1, s2, q0, q1, q2};
  for (int q = 0; q < 6; ++q) {
    rs[t] = vals[q]; __syncthreads();
    for (int off = 128; off > 0; off >>= 1) {
      if (t < off) rs[t] += rs[t + off];
      __syncthreads();
    }
    if (t == 0) atomicAdd(&stats[q], rs[0]);
    __syncthreads();
  }
}

// ---------------- kernel 2: conv1 + BN1 + maxpool(16,4) ----------------
__global__ __launch_bounds__(128)
void k_conv1pool(const float* __restrict__ hm, const float* __restrict__ dets,
                 const float* __restrict__ w1, const float* __restrict__ b1,
                 const float* __restrict__ g1, const float* __restrict__ bb1,
                 const float* __restrict__ stats, float* __restrict__ pooled) {
  const int n = blockIdx.x, t = threadIdx.x;
  const float* H = hm + (size_t)(n / KCLIP) * (RB * RB);
  const float cnt = (float)NB * (float)(RB * RB);
  float sc[3], sh[3];
  #pragma unroll
  for (int c = 0; c < 3; ++c) {
    const float mean = stats[c] / cnt;
    const float var  = stats[3 + c] / cnt - mean * mean;
    const float inv  = rsqrtf(var + EPSBN);
    sc[c] = inv * g1[c];
    sh[c] = bb1[c] - mean * sc[c];
  }
  const float d0 = dets[n*6+0], d1 = dets[n*6+1], d2 = dets[n*6+2], d3 = dets[n*6+3];
  const int left = (int)d0, up = (int)d1, right = (int)(d0 + d2), low = (int)(d1 + d3);
  float W[3][9];
  #pragma unroll
  for (int oc = 0; oc < 3; ++oc)
    #pragma unroll
    for (int k = 0; k < 9; ++k) W[oc][k] = w1[oc*9+k];
  const float cb0 = b1[0], cb1 = b1[1], cb2 = b1[2];

  __shared__ float M[128][97];           // [col][ch*32 + row_group]
  const int c = t;
  for (int gy = 0; gy < 32; ++gy) {
    float m0 = NEGINF, m1 = NEGINF, m2 = NEGINF;
    #pragma unroll
    for (int q = 0; q < 4; ++q) {
      const int r = gy * 4 + q;
      float a0 = cb0, a1 = cb1, a2 = cb2;
      #pragma unroll
      for (int dr = 0; dr < 3; ++dr) {
        const int rr = r + dr - 1;
        const bool rok = (rr >= 0) && (rr < RB) && (rr >= up) && (rr < low);
        #pragma unroll
        for (int dc = 0; dc < 3; ++dc) {
          const int cc = c + dc - 1;
          float v = 0.0f;
          if (rok && cc >= 0 && cc < RB && cc >= left && cc < right) v = H[rr*RB+cc];
          a0 = fmaf(W[0][dr*3+dc], v, a0);
          a1 = fmaf(W[1][dr*3+dc], v, a1);
          a2 = fmaf(W[2][dr*3+dc], v, a2);
        }
      }
      m0 = fmaxf(m0, fmaxf(a0, 0.f) * sc[0] + sh[0]);
      m1 = fmaxf(m1, fmaxf(a1, 0.f) * sc[1] + sh[1]);
      m2 = fmaxf(m2, fmaxf(a2, 0.f) * sc[2] + sh[2]);
    }
    M[c][gy] = m0; M[c][32 + gy] = m1; M[c][64 + gy] = m2;
  }
  __syncthreads();
  for (int j = t; j < NCH*PO*PO; j += 128) {
    const int ch = j / (PO*PO), rem = j % (PO*PO), py = rem / PO, px = rem % PO;
    float m = NEGINF;
    for (int cc = 4*px; cc < 4*px + 16; ++cc)
      #pragma unroll
      for (int g = 0; g < 4; ++g)
        m = fmaxf(m, M[cc][ch*32 + py + g]);
    pooled[(size_t)n * (NCH*PO*PO) + j] = m;
  }
}

// ---------------- kernel 3: conv2 + bias + ReLU, BN2 stats, padded store ----
__global__ __launch_bounds__(256)
void k_conv2(const float* __restrict__ pooled, const float* __restrict__ w2,
             const float* __restrict__ b2, float* __restrict__ stats,
             float* __restrict__ act) {
  const int n = blockIdx.x, t = threadIdx.x;
  __shared__ float P[NCH*PO*PO];         // 2523 floats
  __shared__ float ws2[81];
  __shared__ float bs2[3];
  for (int j = t; j < NCH*PO*PO; j += 256) P[j] = pooled[(size_t)n*(NCH*PO*PO) + j];
  if (t < 81) ws2[t] = w2[t];
  if (t < 3)  bs2[t] = b2[t];
  __syncthreads();
  float ls[3] = {0,0,0}, lq[3] = {0,0,0};
  for (int j = t; j < KP; j += 256) {
    if (j < FDIM) {
      const int c = j / (QO*QO), rem = j % (QO*QO), py = rem / QO, px = rem % QO;
      float acc = bs2[c];
      #pragma unroll
      for (int ic = 0; ic < 3; ++ic)
        #pragma unroll
        for (int dy = 0; dy < 3; ++dy)
          #pragma unroll
          for (int dx = 0; dx < 3; ++dx)
            acc = fmaf(ws2[((c*3+ic)*3+dy)*3+dx],
                       P[ic*(PO*PO) + (py+dy)*PO + (px+dx)], acc);
      acc = fmaxf(acc, 0.f);
      act[(size_t)n * KP + j] = acc;
      ls[c] += acc; lq[c] += acc * acc;
    } else {
      act[(size_t)n * KP + j] = 0.0f;    // zero pad -> no K tail in GEMM
    }
  }
  __shared__ float rs[256];
  float vals[6] = {ls[0], ls[1], ls[2], lq[0], lq[1], lq[2]};
  for (int q = 0; q < 6; ++q) {
    rs[t] = vals[q]; __syncthreads();
    for (int off = 128; off > 0; off >>= 1) {
      if (t < off) rs[t] += rs[t + off];
      __syncthreads();
    }
    if (t == 0) atomicAdd(&stats[q], rs[0]);
    __syncthreads();
  }
}

// ---------------- kernel 4: fold BN2 into FC weights/bias ----------------
__global__ __launch_bounds__(256)
void k_prep(const float* __restrict__ fcw, const float* __restrict__ fcb,
            const float* __restrict__ g2, const float* __restrict__ bb2,
            const float* __restrict__ stats, float* __restrict__ weff,
            float* __restrict__ beff) {
  const int j = blockIdx.x * 256 + threadIdx.x;
  if (j >= NOUT) return;
  const float cnt = (float)NB * (float)(QO*QO);
  float s[3], tt[3];
  #pragma unroll
  for (int c = 0; c < 3; ++c) {
    const float mean = stats[c] / cnt;
    const float var  = stats[3 + c] / cnt - mean * mean;
    const float inv  = rsqrtf(var + EPSBN);
    s[c]  = inv * g2[c];
    tt[c] = bb2[c] - mean * s[c];
  }
  float bacc = fcb[j];
  for (int i = 0; i < FDIM; ++i) {
    const int c = (i >= 2*QO*QO) ? 2 : ((i >= QO*QO) ? 1 : 0);
    const float w = fcw[(size_t)i * NOUT + j];
    weff[(size_t)i * NOUT + j] = w * s[c];
    bacc = fmaf(tt[c], w, bacc);
  }
  for (int i = FDIM; i < KP; ++i) weff[(size_t)i * NOUT + j] = 0.0f;
  beff[j] = bacc;
}

// ---------------- kernel 5: f32 WMMA GEMM  out = act @ weff + beff ----------
// M=6400, N=512, K=2192.  Block 256 threads = 8 waves; tile 128x64;
// wave tile 32x32 = four v_wmma_f32_16x16x4_f32 accumulators.
// Double-buffered LDS: prefetch tile kt+1 into registers while WMMAs consume
// tile kt, commit to the other buffer, single barrier per K-step.
__global__ __launch_bounds__(256)
void k_gemm(const float* __restrict__ A, const float* __restrict__ Bm,
            const float* __restrict__ beff, float* __restrict__ C) {
  __shared__ float As[2][128][17];       // [buf][m][k], pad 17 -> conflict-free gather
  __shared__ float Bs[2][16][66];        // [buf][k][n]
  const int t    = threadIdx.x;
  const int m0   = blockIdx.x * 128;
  const int n0   = blockIdx.y * 64;
  const int lane = t & 31, w = t >> 5;
  const int wm   = (w & 3) * 32;         // 4 waves along M
  const int wn   = (w >> 2) * 32;        // 2 waves along N
  const int half = lane >> 4, lm = lane & 15;

  // per-thread staging coordinates (two A quads + one B quad per thread)
  const int am0 = t >> 2,        ak = (t & 3) * 4;   // A rows 0..63
  const int am1 = 64 + (t >> 2);                     // A rows 64..127
  const int bk  = t >> 4,        bn = (t & 15) * 4;  // B

  v8f acc[2][2] = {};
  const int NT = KP / 16;                // 137 K-tiles

  // preload tile 0 into buffer 0
  {
    const float4 v0 = *(const float4*)(A  + (size_t)(m0 + am0) * KP + ak);
    const float4 v1 = *(const float4*)(A  + (size_t)(m0 + am1) * KP + ak);
    const float4 vb = *(const float4*)(Bm + (size_t)bk * NOUT + n0 + bn);
    As[0][am0][ak+0] = v0.x; As[0][am0][ak+1] = v0.y;
    As[0][am0][ak+2] = v0.z; As[0][am0][ak+3] = v0.w;
    As[0][am1][ak+0] = v1.x; As[0][am1][ak+1] = v1.y;
    As[0][am1][ak+2] = v1.z; As[0][am1][ak+3] = v1.w;
    Bs[0][bk][bn+0]  = vb.x; Bs[0][bk][bn+1]  = vb.y;
    Bs[0][bk][bn+2]  = vb.z; Bs[0][bk][bn+3]  = vb.w;
  }
  __syncthreads();

  for (int kt = 0; kt < NT; ++kt) {
    const int cur = kt & 1, nxt = cur ^ 1;
    const bool more = (kt + 1) < NT;     // wave-uniform -> EXEC stays all-1s
    float4 v0, v1, vb;
    if (more) {                          // prefetch next tile into registers
      const int k0 = (kt + 1) * 16;
      v0 = *(const float4*)(A  + (size_t)(m0 + am0) * KP + k0 + ak);
      v1 = *(const float4*)(A  + (size_t)(m0 + am1) * KP + k0 + ak);
      vb = *(const float4*)(Bm + (size_t)(k0 + bk) * NOUT + n0 + bn);
    }
    #pragma unroll
    for (int kk = 0; kk < 4; ++kk) {
      const int kb = kk * 4 + half * 2;  // lane's K pair within the k=4 step
      v2f a0, a1, b0, b1;
      a0.x = As[cur][wm + lm][kb];       a0.y = As[cur][wm + lm][kb + 1];
      a1.x = As[cur][wm + 16 + lm][kb];  a1.y = As[cur][wm + 16 + lm][kb + 1];
      b0.x = Bs[cur][kb][wn + lm];       b0.y = Bs[cur][kb + 1][wn + lm];
      b1.x = Bs[cur][kb][wn + 16 + lm];  b1.y = Bs[cur][kb + 1][wn + 16 + lm];
      acc[0][0] = __builtin_amdgcn_wmma_f32_16x16x4_f32(false, a0, false, b0,
                    (short)0, acc[0][0], false, false);
      acc[0][1] = __builtin_amdgcn_wmma_f32_16x16x4_f32(false, a0, false, b1,
                    (short)0, acc[0][1], false, false);
      acc[1][0] = __builtin_amdgcn_wmma_f32_16x16x4_f32(false, a1, false, b0,
                    (short)0, acc[1][0], false, false);
      acc[1][1] = __builtin_amdgcn_wmma_f32_16x16x4_f32(false, a1, false, b1,
                    (short)0, acc[1][1], false, false);
    }
    if (more) {                          // commit prefetched tile to other buffer
      As[nxt][am0][ak+0] = v0.x; As[nxt][am0][ak+1] = v0.y;
      As[nxt][am0][ak+2] = v0.z; As[nxt][am0][ak+3] = v0.w;
      As[nxt][am1][ak+0] = v1.x; As[nxt][am1][ak+1] = v1.y;
      As[nxt][am1][ak+2] = v1.z; As[nxt][am1][ak+3] = v1.w;
      Bs[nxt][bk][bn+0]  = vb.x; Bs[nxt][bk][bn+1]  = vb.y;
      Bs[nxt][bk][bn+2]  = vb.z; Bs[nxt][bk][bn+3]  = vb.w;
    }
    __syncthreads();
  }

  // epilogue: C layout lane l, vgpr v -> (M = v + 8*(l/16), N = l%16)
  #pragma unroll
  for (int tm = 0; tm < 2; ++tm)
    #pragma unroll
    for (int tn = 0; tn < 2; ++tn) {
      const int col = n0 + wn + tn * 16 + lm;
      const float bb = beff[col];
      #pragma unroll
      for (int v = 0; v < 8; ++v) {
        const int row = m0 + wm + tm * 16 + v + 8 * half;
        C[(size_t)row * NOUT + col] = acc[tm][tn][v] + bb;
      }
    }
}

// ---------------- host launch ----------------
extern "C" void kernel_launch(void* const* d_in, const int* in_sizes, int n_in,
                              void* d_out, int out_size, void* d_ws, size_t ws_size,
                              hipStream_t stream) {
  const float* heat = (const float*)d_in[0];
  const float* dets = (const float*)d_in[1];
  const float* c1w  = (const float*)d_in[2];
  const float* c1b  = (const float*)d_in[3];
  const float* g1   = (const float*)d_in[4];
  const float* bb1  = (const float*)d_in[5];
  const float* c2w  = (const float*)d_in[6];
  const float* c2b  = (const float*)d_in[7];
  const float* g2   = (const float*)d_in[8];
  const float* bb2  = (const float*)d_in[9];
  const float* fcw  = (const float*)d_in[10];
  const float* fcb  = (const float*)d_in[11];
  float* ws  = (float*)d_ws;
  float* out = (float*)d_out;

  float* stats1 = ws + O_STATS1;
  float* stats2 = ws + O_STATS2;
  float* beff   = ws + O_BEFF;
  float* pooled = ws + O_POOLED;
  float* act    = ws + O_ACT;
  float* weff   = ws + O_WEFF;

  k_init<<<1, 32, 0, stream>>>(ws);
  k_stats1<<<NB, 256, 0, stream>>>(heat, dets, c1w, c1b, stats1);
  k_conv1pool<<<NB, 128, 0, stream>>>(heat, dets, c1w, c1b, g1, bb1, stats1, pooled);
  k_conv2<<<NB, 256, 0, stream>>>(pooled, c2w, c2b, stats2, act);
  k_prep<<<2, 256, 0, stream>>>(fcw, fcb, g2, bb2, stats2, weff, beff);
  k_gemm<<<dim3(NB/128, NOUT/64), 256, 0, stream>>>(act, weff, beff, out);
}